// AngularDensity_34797825032453
// MI455X (gfx1250) — compile-verified
//
#include <hip/hip_runtime.h>
#include <hip/hip_bf16.h>

#define NBATCH   16
#define NUMATOM  512
#define NEIGH    48
#define NPAIR    (NUMATOM * NEIGH)        /* 24576 per batch    */
#define NPTOT    (NBATCH * NPAIR)         /* 393216 total pairs */
#define TOTNATOM (NBATCH * NUMATOM)       /* 8192 atoms         */
#define NTYPE    4
#define NWAVE    8

typedef __attribute__((ext_vector_type(2))) float v2f;
typedef __attribute__((ext_vector_type(8))) float v8f;

// ---------------------------------------------------------------------------
// Pass 1: count pairs per center atom
// ---------------------------------------------------------------------------
__global__ __launch_bounds__(256) void count_kernel(const int* __restrict__ atom_index,
                                                    int* __restrict__ count) {
    int gp = blockIdx.x * blockDim.x + threadIdx.x;
    if (gp >= NPTOT) return;
    int b = gp / NPAIR;
    int p = gp - b * NPAIR;
    int center = atom_index[(size_t)b * 2 * NPAIR + p] + b * NUMATOM;
    atomicAdd(&count[center], 1);
}

// ---------------------------------------------------------------------------
// Pass 2: exclusive scan of the 8192 counts (single block, 256 threads,
// 32 elements per thread, Hillis-Steele over the 256 partials in LDS)
// ---------------------------------------------------------------------------
__global__ __launch_bounds__(256) void scan_kernel(const int* __restrict__ count,
                                                   int* __restrict__ offsets,
                                                   int* __restrict__ cursor) {
    __shared__ int part[256];
    int t = threadIdx.x;
    int base = t * 32;
    int s = 0;
    for (int i = 0; i < 32; ++i) s += count[base + i];
    part[t] = s;
    __syncthreads();
    for (int off = 1; off < 256; off <<= 1) {
        int v = (t >= off) ? part[t - off] : 0;
        __syncthreads();
        part[t] += v;
        __syncthreads();
    }
    int run = (t == 0) ? 0 : part[t - 1];
    for (int i = 0; i < 32; ++i) {
        offsets[base + i] = run;
        cursor[base + i]  = run;
        run += count[base + i];
    }
}

// ---------------------------------------------------------------------------
// Pass 3: scatter pair ids into per-atom buckets
// ---------------------------------------------------------------------------
__global__ __launch_bounds__(256) void scatter_kernel(const int* __restrict__ atom_index,
                                                      int* __restrict__ cursor,
                                                      int* __restrict__ pairlist) {
    int gp = blockIdx.x * blockDim.x + threadIdx.x;
    if (gp >= NPTOT) return;
    int b = gp / NPAIR;
    int p = gp - b * NPAIR;
    int center = atom_index[(size_t)b * 2 * NPAIR + p] + b * NUMATOM;
    int pos = atomicAdd(&cursor[center], 1);
    pairlist[pos] = gp;
}

// ---------------------------------------------------------------------------
// Pass 4: one wave32 per atom. All pairs of an atom share one species
// (neigh_sp = species[center]), so one type-packed B operand suffices:
//   pd(16x16) += Ang(16x4) x G(4x16),  G columns 8*(t&1)..+7 carry the radial
// basis (masked by tail/validity factor), the other 8 columns stay zero
// (zeroed once before the loop, never rewritten). 16 pairs staged per round
// by lanes 0..15 (branch-free, clamped index), then 4x V_WMMA_F32_16X16X4_F32.
// Afterwards: square, reduce angular orders (rows 0-2 -> seg0, rows 3-11 ->
// seg1, seg2 == 0), combine C halves with a cross-lane shuffle, write 96
// outputs per atom (non-matching type blocks are structural zeros).
// ---------------------------------------------------------------------------
__global__ __launch_bounds__(64) void density_kernel(
        const float* __restrict__ coords,
        const int*   __restrict__ atom_index,
        const float* __restrict__ shifts,
        const int*   __restrict__ species,
        const float* __restrict__ rs,
        const float* __restrict__ inta,
        const int*   __restrict__ offsets,
        const int*   __restrict__ count,
        const int*   __restrict__ pairlist,
        float*       __restrict__ out) {
    __shared__ float stage[2][16][32];   // [wave][pair-in-round][ang16 | Bcols16]

    const int lane = threadIdx.x & 31;
    const int wv   = threadIdx.x >> 5;
    const int atom = blockIdx.x * 2 + wv;          // 4096 blocks * 2 waves = 8192 atoms

    const int start = offsets[atom];
    const int cnt   = count[atom];
    const int t     = species[atom];               // species of the center atom
    const int gcol  = 16 + (t & 1) * 8;            // B column block for this type

    // per-atom radial parameters (uniform across the bucket)
    float rsv[NWAVE], negia[NWAVE];
    #pragma unroll
    for (int w = 0; w < NWAVE; ++w) {
        rsv[w]   = rs[t * NWAVE + w];
        negia[w] = -10.0f * inta[t * NWAVE + w];
    }

    // zero the full records once: pad rows 12-15 and the off-parity B block
    // stay zero for the whole bucket.
    if (lane < 16) {
        float* sp = &stage[wv][lane][0];
        #pragma unroll
        for (int m = 0; m < 32; ++m) sp[m] = 0.0f;
    }

    v8f acc = {};

    const int nround = (cnt + 15) >> 4;
    for (int rd = 0; rd < nround; ++rd) {
        __builtin_prefetch(pairlist + start + (rd + 1) * 16, 0, 1);

        // ---- stage 16 pairs (lanes 0..15), branch-free --------------------
        if (lane < 16) {
            int idx     = start + rd * 16 + lane;
            bool active = idx < start + cnt;
            int idxc    = active ? idx : start;        // clamp: duplicate pair 0
            int gp      = pairlist[idxc];
            int b       = gp / NPAIR;
            int p       = gp - b * NPAIR;
            const int* ai = atom_index + (size_t)b * 2 * NPAIR;
            int i = ai[p] + b * NUMATOM;               // center (global) == atom
            int j = ai[NPAIR + p] + b * NUMATOM;       // neighbor (global)
            const float* sh = shifts + (size_t)gp * 3;
            float sx = sh[0], sy = sh[1], sz = sh[2];
            bool valid = (sx > -1e9f) && (sy > -1e9f) && (sz > -1e9f);
            float fac = (active && valid) ? 1.0f : 0.0f;

            float dx = coords[3 * i + 0] - coords[3 * j + 0] + sx;
            float dy = coords[3 * i + 1] - coords[3 * j + 1] + sy;
            float dz = coords[3 * i + 2] - coords[3 * j + 2] + sz;
            float r   = sqrtf(dx * dx + dy * dy + dz * dz);
            float inv = 1.0f / r;
            float u[3] = {dx * inv, dy * inv, dz * inv};

            float* sp = &stage[wv][lane][0];           // 128B-aligned per lane
            sp[0] = u[0]; sp[1] = u[1]; sp[2] = u[2];
            #pragma unroll
            for (int a2 = 0; a2 < 3; ++a2)
                #pragma unroll
                for (int b2 = 0; b2 < 3; ++b2)
                    sp[3 + a2 * 3 + b2] = u[a2] * u[b2];
            #pragma unroll
            for (int w = 0; w < NWAVE; ++w) {
                float d = r - rsv[w];
                sp[gcol + w] = __expf(negia[w] * d * d) * fac;
            }
        }
        __asm volatile("s_wait_dscnt 0" ::: "memory");

        // ---- four K=4 WMMAs per round -------------------------------------
        // A (16x4 f32): lanes 0-15 hold M=lane, VGPR0=K0, VGPR1=K1;
        //               lanes 16-31 hold K2/K3.  B (4x16) mirrors the K split.
        int m  = lane & 15;
        int k0 = (lane >> 4) * 2;
        #pragma unroll
        for (int c = 0; c < 4; ++c) {
            int kp = c * 4 + k0;
            v2f A = { stage[wv][kp][m],      stage[wv][kp + 1][m]      };
            v2f B = { stage[wv][kp][16 + m], stage[wv][kp + 1][16 + m] };
            acc = __builtin_amdgcn_wmma_f32_16x16x4_f32(
                      false, A, false, B, (short)0, acc, false, false);
        }
        __asm volatile("s_wait_dscnt 0" ::: "memory");
    }

    // ---- square + angular-order reduction --------------------------------
    // C layout: lane holds N = lane%16; VGPR v is row M = v + 8*(lane>=16).
    float s0 = 0.f, s1 = 0.f, shi = 0.f;
    #pragma unroll
    for (int v = 0; v < 3; ++v) s0  += acc[v] * acc[v];   // rows 0-2
    #pragma unroll
    for (int v = 3; v < 8; ++v) s1  += acc[v] * acc[v];   // rows 3-7
    #pragma unroll
    for (int v = 0; v < 4; ++v) shi += acc[v] * acc[v];   // rows 8-11 (hi half)

    float hi = __shfl(shi, (lane & 15) + 16, 32);

    if (lane < 16) {
        int w  = lane & 7;
        int cb = lane >> 3;                 // C column block (0 or 1)
        int tt1 = (t & 2) | cb;             // type mapped to this C half
        int tt2 = tt1 ^ 2;                  // type in the other half: all zero
        size_t base1 = (size_t)atom * 96 + (size_t)tt1 * 24;
        out[base1 + w]      = s0;           // zero unless cb == (t&1)
        out[base1 + 8 + w]  = s1 + hi;
        out[base1 + 16 + w] = 0.0f;         // order-2 segment is structurally zero
        size_t base2 = (size_t)atom * 96 + (size_t)tt2 * 24;
        out[base2 + w]      = 0.0f;
        out[base2 + 8 + w]  = 0.0f;
        out[base2 + 16 + w] = 0.0f;
    }
}

// ---------------------------------------------------------------------------
extern "C" void kernel_launch(void* const* d_in, const int* in_sizes, int n_in,
                              void* d_out, int out_size, void* d_ws, size_t ws_size,
                              hipStream_t stream) {
    const float* coords     = (const float*)d_in[0];
    /* d_in[1] = numatoms (unused by the forward math) */
    const int*   atom_index = (const int*)  d_in[2];
    const float* shifts     = (const float*)d_in[3];
    const int*   species    = (const int*)  d_in[4];
    const float* rs         = (const float*)d_in[5];
    const float* inta       = (const float*)d_in[6];
    float*       out        = (float*)d_out;

    int* count    = (int*)d_ws;            // 8192
    int* offsets  = count + TOTNATOM;      // 8192
    int* cursor   = offsets + TOTNATOM;    // 8192
    int* pairlist = cursor + TOTNATOM;     // 393216

    hipMemsetAsync(count, 0, TOTNATOM * sizeof(int), stream);

    count_kernel  <<<(NPTOT + 255) / 256, 256, 0, stream>>>(atom_index, count);
    scan_kernel   <<<1, 256, 0, stream>>>(count, offsets, cursor);
    scatter_kernel<<<(NPTOT + 255) / 256, 256, 0, stream>>>(atom_index, cursor, pairlist);
    density_kernel<<<TOTNATOM / 2, 64, 0, stream>>>(coords, atom_index, shifts, species,
                                                    rs, inta, offsets, count, pairlist, out);
}